// GraphConv_21766894256813
// MI455X (gfx1250) — compile-verified
//
#include <hip/hip_runtime.h>

#define D_   128
#define C_   64
#define OUT_ 128
#define KTOT 320   // D + C + D

typedef __attribute__((ext_vector_type(2))) float v2f;
typedef __attribute__((ext_vector_type(8))) float v8f;

// ---------------- zero workspace (U and W accumulators) ----------------
__global__ void zero_ws(float* __restrict__ p, long n) {
  long i = ((long)blockIdx.x * blockDim.x + threadIdx.x) * 4;
  if (i + 3 < n) {
    *(float4*)(p + i) = make_float4(0.f, 0.f, 0.f, 0.f);
  } else if (i < n) {
    for (long j = i; j < n; ++j) p[j] = 0.f;
  }
}

// ---------------- pack combined transposed weight, K-pair interleaved ----------------
// Logical B[k][n], k in [0,320), n in [0,128):
//   k 0..127   : linear_w[n][k]                                (U part)
//   k 128..191 : M2[r][n] = ccle[r,:] . linear_w[n,128:192]    (relation part)
//   k 192..319 : self_w[n][k-192]                              (self loop)
// Stored as Bp[(k>>1)*256 + n*2 + (k&1)] so each lane's {K=kk, K=kk+1} pair is
// one aligned 8-byte load in the WMMA loop.
__global__ void pack_B(const float* __restrict__ linear_w,
                       const float* __restrict__ self_w,
                       const float* __restrict__ ccle,
                       float* __restrict__ Bp) {
  int i = blockIdx.x * blockDim.x + threadIdx.x;
  if (i >= KTOT * OUT_) return;
  int k = i >> 7;          // 0..319
  int n = i & (OUT_ - 1);  // 0..127
  float v;
  if (k < D_) {
    v = linear_w[n * (D_ + C_) + k];
  } else if (k < D_ + C_) {
    int r = k - D_;
    float acc = 0.f;
#pragma unroll
    for (int c = 0; c < C_; ++c)
      acc += ccle[r * C_ + c] * linear_w[n * (D_ + C_) + D_ + c];
    v = acc;
  } else {
    v = self_w[n * D_ + (k - (D_ + C_))];
  }
  Bp[(k >> 1) * (2 * OUT_) + n * 2 + (k & 1)] = v;
}

// ---------------- edge scatter: U[dst,:] += w*x[src,:],  W[dst,rel] += w ----------------
// One wave handles 32 consecutive edges: metadata loads are coalesced (one edge
// per lane), then broadcast per-edge via uniform-index __shfl (v_readlane).
// x (51MB), U (51MB), W (26MB) are all L2-resident on MI455X (192MB L2), so the
// native global_atomic_add_f32 traffic stays on-chip.
__global__ void edge_scatter(const float* __restrict__ x,
                             const int* __restrict__ node_in,
                             const int* __restrict__ node_out,
                             const int* __restrict__ relation,
                             const float* __restrict__ ew,
                             float* __restrict__ U,
                             float* __restrict__ Wacc,
                             int E) {
  const int lane = threadIdx.x & 31;
  const long wave = ((long)blockIdx.x * blockDim.x + threadIdx.x) >> 5;
  const long base = wave * 32;
  if (base >= E) return;
  const int cnt = (E - base) < 32 ? (int)(E - base) : 32;

  int dstv = 0, srcv = 0, relv = 0;
  float wv = 0.f;
  if (lane < cnt) {
    long e = base + lane;
    dstv = node_in[e];
    srcv = node_out[e];
    relv = relation[e];
    wv   = ew[e];
  }

  for (int j = 0; j < cnt; ++j) {
    int   dst = __shfl(dstv, j);
    int   src = __shfl(srcv, j);
    int   rel = __shfl(relv, j);
    float w   = __shfl(wv, j);

    float4 xv = *((const float4*)x + (long)src * (D_ / 4) + lane);  // coalesced 512B row
    float* urow = U + (long)dst * D_ + lane * 4;
    unsafeAtomicAdd(urow + 0, w * xv.x);
    unsafeAtomicAdd(urow + 1, w * xv.y);
    unsafeAtomicAdd(urow + 2, w * xv.z);
    unsafeAtomicAdd(urow + 3, w * xv.w);
    if (lane == 0) unsafeAtomicAdd(Wacc + (long)dst * C_ + rel, w);
  }
}

// ---------------- fused GEMM + bias + ReLU using V_WMMA_F32_16X16X4_F32 ----------------
// out[m,n] = relu( [U | W | x][m,:] @ B[:, n] + linear_b[n] + self_b[n] )
// Block: 256 threads = 8 waves. Block owns 16 rows (A tile staged in LDS),
// wave nt owns columns [16*nt, 16*nt+16). K = 320 -> 80 wmma ops per wave.
__global__ __launch_bounds__(256)
void gemm_epilogue(const float* __restrict__ U,
                   const float* __restrict__ Wacc,
                   const float* __restrict__ x,
                   const float* __restrict__ Bp,
                   const float* __restrict__ linear_b,
                   const float* __restrict__ self_b,
                   float* __restrict__ out,
                   int N) {
  __shared__ float Atile[16][KTOT + 4];  // stride 324: 324%64==4 -> conflict-free column reads
  const int tid = threadIdx.x;
  const int mbase = blockIdx.x * 16;

  // cooperative staging of the 16x320 A tile (rows clamped for the tail tile)
  for (int i = tid; i < 16 * D_; i += 256) {
    int r = i >> 7, c = i & (D_ - 1);
    int m = mbase + r; if (m > N - 1) m = N - 1;
    Atile[r][c] = U[(long)m * D_ + c];
  }
  for (int i = tid; i < 16 * C_; i += 256) {
    int r = i >> 6, c = i & (C_ - 1);
    int m = mbase + r; if (m > N - 1) m = N - 1;
    Atile[r][D_ + c] = Wacc[(long)m * C_ + c];
  }
  for (int i = tid; i < 16 * D_; i += 256) {
    int r = i >> 7, c = i & (D_ - 1);
    int m = mbase + r; if (m > N - 1) m = N - 1;
    Atile[r][D_ + C_ + c] = x[(long)m * D_ + c];
  }
  __syncthreads();

  const int lane  = tid & 31;
  const int half  = lane >> 4;   // 0: lower K/M half, 1: upper half
  const int idx16 = lane & 15;   // M index for A, N index for B/C/D
  const int nbase = (tid >> 5) * 16;
  const int n     = nbase + idx16;

  // loop-invariant part of the packed-B address: row (kk>>1) = k/2 + half
  const float* bptr = Bp + (long)half * (2 * OUT_) + (n << 1);

  v8f acc = {};
#pragma unroll 8
  for (int k = 0; k < KTOT; k += 4) {
    const int kk = k + 2 * half;
    // A fragment (16x4 f32): lane<16 holds {K=k,k+1} of row M=lane; lane>=16 holds {K=k+2,k+3}
    v2f a = *(const v2f*)&Atile[idx16][kk];
    // B fragment (4x16 f32): one b64 load gives {B[kk][n], B[kk+1][n]}
    v2f b = *(const v2f*)(bptr + (k >> 1) * (2 * OUT_));
    acc = __builtin_amdgcn_wmma_f32_16x16x4_f32(false, a, false, b, (short)0, acc,
                                                false, false);
  }

  const float bias = linear_b[n] + self_b[n];
  if (mbase + 16 <= N) {
    // full tile: straight-line stores, one address + immediate offsets (i*512B)
    float* orow = out + (long)(mbase + half * 8) * OUT_ + n;
#pragma unroll
    for (int i = 0; i < 8; ++i) {
      float v = acc[i] + bias;
      orow[i * OUT_] = v > 0.f ? v : 0.f;
    }
  } else {
#pragma unroll
    for (int i = 0; i < 8; ++i) {
      int m = mbase + i + half * 8;  // C/D layout: VGPR i -> M=i (lanes 0-15), M=8+i (lanes 16-31)
      if (m < N) {
        float v = acc[i] + bias;
        out[(long)m * OUT_ + n] = v > 0.f ? v : 0.f;
      }
    }
  }
}

extern "C" void kernel_launch(void* const* d_in, const int* in_sizes, int n_in,
                              void* d_out, int out_size, void* d_ws, size_t ws_size,
                              hipStream_t stream) {
  const float* x        = (const float*)d_in[0];
  const int*   node_in  = (const int*)d_in[1];
  const int*   node_out = (const int*)d_in[2];
  const int*   relation = (const int*)d_in[3];
  const float* ew       = (const float*)d_in[4];
  const float* ccle     = (const float*)d_in[5];
  const float* linear_w = (const float*)d_in[6];
  const float* linear_b = (const float*)d_in[7];
  const float* self_w   = (const float*)d_in[8];
  const float* self_b   = (const float*)d_in[9];

  const int N = in_sizes[0] / D_;
  const int E = in_sizes[1];

  float* U    = (float*)d_ws;               // [N,128] fp32
  float* Wacc = U + (size_t)N * D_;         // [N, 64] fp32
  float* Bp   = Wacc + (size_t)N * C_;      // [160][128][2] fp32 (K-pair interleaved)

  // 1) zero accumulators (must happen every call: graph replays re-accumulate)
  long nzero = (long)N * (D_ + C_);
  long zthreads = (nzero + 3) / 4;
  zero_ws<<<(int)((zthreads + 255) / 256), 256, 0, stream>>>(U, nzero);

  // 2) build the packed transposed weight matrix (includes tiny ccle@linear_w GEMM)
  pack_B<<<(KTOT * OUT_ + 255) / 256, 256, 0, stream>>>(linear_w, self_w, ccle, Bp);

  // 3) edge scatter: one wave per 32 edges
  long waves = ((long)E + 31) / 32;
  long blocks = (waves + 7) / 8;
  edge_scatter<<<(int)blocks, 256, 0, stream>>>(x, node_in, node_out, relation, ew,
                                                U, Wacc, E);

  // 4) fused WMMA GEMM + bias + ReLU
  int mtiles = (N + 15) / 16;
  gemm_epilogue<<<mtiles, 256, 0, stream>>>(U, Wacc, x, Bp, linear_b, self_b,
                                            (float*)d_out, N);
}